// ReBasedAttention_72473278153328
// MI455X (gfx1250) — compile-verified
//
#include <hip/hip_runtime.h>
#include <math.h>

// MI455X / gfx1250, wave32.
//  - fp32 WMMA (V_WMMA_F32_16X16X4_F32) for all matrix math
//  - GEMM tiles staged via async global->LDS b128 copies (ASYNCcnt)
//  - attention Q/K/V tiles staged via Tensor Data Mover (TENSORcnt),
//    with TDM hardware LDS padding providing the bank-conflict-free pitch
typedef __attribute__((ext_vector_type(2))) float v2f;
typedef __attribute__((ext_vector_type(8))) float v8f;
typedef __attribute__((ext_vector_type(4))) unsigned v4u;
typedef __attribute__((ext_vector_type(8))) unsigned v8u;

static __device__ __forceinline__ v8f wmma4(v2f a, v2f b, v8f c) {
  // (neg_a, A, neg_b, B, c_mod, C, reuse_a, reuse_b)
  return __builtin_amdgcn_wmma_f32_16x16x4_f32(false, a, false, b, (short)0, c,
                                               false, false);
}

// Async copy of 16B/lane into LDS, GVS addressing (SGPR base + u32 offset).
static __device__ __forceinline__ void async_b128(unsigned lds_byte,
                                                  const float* sbase,
                                                  unsigned glb_byte) {
  asm volatile("global_load_async_to_lds_b128 %0, %1, %2"
               :
               : "v"(lds_byte), "v"(glb_byte), "s"(sbase)
               : "memory");
}
static __device__ __forceinline__ void wait_async0() {
  asm volatile("s_wait_asynccnt 0x0" ::: "memory");
}
static __device__ __forceinline__ unsigned lds_off(const void* p) {
  return (unsigned)(unsigned long long)p;  // low 32 bits = LDS byte offset
}

// ---------------------------------------------------------------------------
// TDM: one-instruction DMA of a 2D f32 tile (tile_w x tile_h elements, row
// stride stride_elems) from global to LDS, optional LDS padding.
// D# layout per CDNA5 ISA 8.3/8.4; 2-group form (groups 2/3 NULL -> 2D).
// ---------------------------------------------------------------------------
static __device__ __forceinline__ void tdm_load_2d(
    unsigned lds_byte, const void* gaddr, unsigned tile_w, unsigned tile_h,
    unsigned stride_elems, unsigned pad_en, unsigned pad_interval_code,
    unsigned pad_amount_code) {
  const unsigned long long ga = (unsigned long long)gaddr;
  v4u g0;
  g0[0] = 1u;  // count=1 valid descriptor; no gather
  g0[1] = lds_byte;
  g0[2] = (unsigned)(ga & 0xFFFFFFFFu);                        // addr[31:0]
  g0[3] = (unsigned)((ga >> 32) & 0x01FFFFFFu) | (2u << 30);   // addr[56:32], type=2
  v8u g1;
  g1[0] = (2u << 16)  // data_size = 4B
          | (pad_en << 20) | (pad_interval_code << 22) | (pad_amount_code << 25);
  g1[1] = (tile_w & 0xFFFFu) << 16;                       // tensor_dim0 = tile_w
  g1[2] = ((tile_h & 0xFFFFu) << 16);                     // tensor_dim1 = tile_h
  g1[3] = (tile_w & 0xFFFFu) << 16;                       // tile_dim0
  g1[4] = tile_h & 0xFFFFu;                               // tile_dim1 (tile_dim2=0)
  g1[5] = stride_elems;                                   // tensor_dim0_stride lo
  g1[6] = 0u;
  g1[7] = 0u;
  asm volatile("tensor_load_to_lds %0, %1" : : "s"(g0), "s"(g1) : "memory");
}

// ---------------------------------------------------------------------------
// GEMM  C[M,N] = A[M,K] @ B[K,N] + bias[N]  (fp32 row-major)
// 256 thr = 8 waves; wave tile 32x64; block tile 128x128; K staged in 32-wide
// double-buffered LDS chunks via async b128 copies.
// ---------------------------------------------------------------------------
#define GKC 32
#define APITCH 36   // floats; 144B rows: 16B aligned, conflict-free frag reads
#define BPITCH 128  // floats

__global__ __launch_bounds__(256) void gemm_bias_kernel(
    const float* __restrict__ A, const float* __restrict__ B,
    const float* __restrict__ bias, float* __restrict__ C, int M, int N,
    int K) {
  __shared__ float Asb[2][128 * APITCH];  // 2 x 18KB
  __shared__ float Bsb[2][GKC * BPITCH];  // 2 x 16KB

  const int tid = threadIdx.x;
  const int lane = tid & 31;
  const int wave = tid >> 5;
  const int l15 = lane & 15;
  const int hi = lane >> 4;
  const int wm = wave >> 1;  // 0..3
  const int wn = wave & 1;   // 0..1
  const int m0 = blockIdx.x * 128;
  const int n0 = blockIdx.y * 128;

  const unsigned aL[2] = {lds_off(&Asb[0][0]), lds_off(&Asb[1][0])};
  const unsigned bL[2] = {lds_off(&Bsb[0][0]), lds_off(&Bsb[1][0])};

  auto stage = [&](int bufI, int k0) {
#pragma unroll
    for (int i = 0; i < 4; ++i) {
      const int idx = tid + i * 256;  // 0..1023
      const int ar = idx >> 3, ac = idx & 7;  // A: 128 rows x 8 x 16B
      const unsigned ag =
          (unsigned)((((size_t)(m0 + ar)) * K + k0) * 4) + ac * 16;
      async_b128(aL[bufI] + (unsigned)(ar * (APITCH * 4) + ac * 16), A, ag);
      const int br = idx >> 5, bc = idx & 31;  // B: 32 rows x 32 x 16B
      const unsigned bg =
          (unsigned)((((size_t)(k0 + br)) * N + n0) * 4) + bc * 16;
      async_b128(bL[bufI] + (unsigned)(br * (BPITCH * 4) + bc * 16), B, bg);
    }
  };

  v8f acc[2][4];
#pragma unroll
  for (int i = 0; i < 2; ++i)
#pragma unroll
    for (int j = 0; j < 4; ++j) acc[i][j] = (v8f){};

  stage(0, 0);
  const int nChunks = K / GKC;
  for (int c = 0; c < nChunks; ++c) {
    wait_async0();
    __syncthreads();  // staged chunk c visible to all waves
    if (c + 1 < nChunks) stage((c + 1) & 1, (c + 1) * GKC);
    const float* As_ = Asb[c & 1];
    const float* Bs_ = Bsb[c & 1];
#pragma unroll
    for (int kk = 0; kk < GKC; kk += 4) {
      v2f a0 = *(const v2f*)&As_[(wm * 32 + l15) * APITCH + kk + 2 * hi];
      v2f a1 = *(const v2f*)&As_[(wm * 32 + 16 + l15) * APITCH + kk + 2 * hi];
      v2f bf[4];
#pragma unroll
      for (int nt = 0; nt < 4; ++nt) {
        bf[nt].x = Bs_[(kk + 2 * hi) * BPITCH + wn * 64 + nt * 16 + l15];
        bf[nt].y = Bs_[(kk + 2 * hi + 1) * BPITCH + wn * 64 + nt * 16 + l15];
      }
#pragma unroll
      for (int nt = 0; nt < 4; ++nt) {
        acc[0][nt] = wmma4(a0, bf[nt], acc[0][nt]);
        acc[1][nt] = wmma4(a1, bf[nt], acc[1][nt]);
      }
    }
    __syncthreads();  // done reading buffer before it is restaged
  }

#pragma unroll
  for (int mt = 0; mt < 2; ++mt)
#pragma unroll
    for (int nt = 0; nt < 4; ++nt) {
      const int col = n0 + wn * 64 + nt * 16 + l15;
      const float bv = bias[col];
#pragma unroll
      for (int r = 0; r < 8; ++r) {
        const int row = m0 + wm * 32 + mt * 16 + r + 8 * hi;
        C[(size_t)row * N + col] = acc[mt][nt][r] + bv;
      }
    }
}

// ---------------------------------------------------------------------------
// In-place quadratic feature map on q/k slices of qkv.
// ---------------------------------------------------------------------------
__global__ __launch_bounds__(256) void featuremap_kernel(
    float* __restrict__ qkv, const float* __restrict__ gamma_q,
    const float* __restrict__ beta_q, const float* __restrict__ gamma_k,
    const float* __restrict__ beta_k, int C, int threeC) {
  const int row = blockIdx.x;
  const int which = blockIdx.y;  // 0 = q, 1 = k
  float* p = qkv + (size_t)row * threeC + which * C;
  const float* gamma = which ? gamma_k : gamma_q;
  const float* beta = which ? beta_k : beta_q;

  float v[4];
  float s = 0.f, s2 = 0.f;
#pragma unroll
  for (int i = 0; i < 4; ++i) {
    v[i] = p[threadIdx.x + i * 256];
    s += v[i];
    s2 += v[i] * v[i];
  }
#pragma unroll
  for (int off = 16; off > 0; off >>= 1) {
    s += __shfl_down(s, off);
    s2 += __shfl_down(s2, off);
  }
  __shared__ float rs[8], rs2[8];
  __shared__ float stat[2];
  const int wave = threadIdx.x >> 5, lane = threadIdx.x & 31;
  if (lane == 0) {
    rs[wave] = s;
    rs2[wave] = s2;
  }
  __syncthreads();
  if (threadIdx.x == 0) {
    float S = 0.f, S2 = 0.f;
    for (int i = 0; i < 8; ++i) {
      S += rs[i];
      S2 += rs2[i];
    }
    const float mean = S / (float)C;
    const float var = (S2 - (float)C * mean * mean) / (float)(C - 1);
    stat[0] = mean;
    stat[1] = 1.f / (sqrtf(var) + 1e-6f);
  }
  __syncthreads();
  const float mean = stat[0], inv = stat[1];
#pragma unroll
  for (int i = 0; i < 4; ++i) {
    const int c = threadIdx.x + i * 256;
    const float xn = (v[i] - mean) * inv;
    const float t = gamma[c] * xn + beta[c];
    p[c] = t * t;
  }
}

// ---------------------------------------------------------------------------
// Chunked causal linear attention, one block per (b,h), chunk = 64 = D.
// Q/K/V chunk tiles staged by the Tensor Data Mover (double buffered);
// TDM pad feature (4 dwords per 64) creates the 68-float LDS pitch.
//   A = Q K^T (masked -> LDS); z = rowsum(A) + Q.ksum_prev
//   O = (Q S + A V) / (z + 1e-6);  S += K^T V;  ksum += colsum(K)
// ---------------------------------------------------------------------------
#define QPITCH 68  // floats; 272B rows: 16B aligned, bank stride 4

__global__ __launch_bounds__(256) void rebased_attn_kernel(
    const float* __restrict__ qkv, float* __restrict__ out, int T, int C,
    int Hh, int D) {
  __shared__ float S[64][65];
  __shared__ float Am[64][66];
  __shared__ float Qs[2][64 * QPITCH];
  __shared__ float Ks[2][64 * QPITCH];
  __shared__ float Vs[2][64 * QPITCH];
  __shared__ float ksum[64];
  __shared__ float zinv[64];

  const int tid = threadIdx.x;
  const int lane = tid & 31;
  const int wave = tid >> 5;
  const int l15 = lane & 15;
  const int hi = lane >> 4;
  const int bh = blockIdx.x;
  const int b = bh / Hh, h = bh % Hh;
  const int threeC = 3 * C;
  const size_t headOff = (size_t)b * T * threeC + (size_t)h * D;  // floats
  float* Obase = out + (size_t)b * T * C + h * D;                 // stride C

  const unsigned qL[2] = {lds_off(&Qs[0][0]), lds_off(&Qs[1][0])};
  const unsigned kL[2] = {lds_off(&Ks[0][0]), lds_off(&Ks[1][0])};
  const unsigned vL[2] = {lds_off(&Vs[0][0]), lds_off(&Vs[1][0])};

  // TDM stage of the three 64x64 tiles for chunk at c0 (wave 0 only).
  // pad: interval code 5 = every 64 dwords, amount code 3 = 4 dwords -> 68.
  auto stageQKV = [&](int bufI, int c0) {
    const float* g = qkv + headOff + (size_t)c0 * threeC;
    tdm_load_2d(qL[bufI], g, 64u, 64u, (unsigned)threeC, 1u, 5u, 3u);
    tdm_load_2d(kL[bufI], g + C, 64u, 64u, (unsigned)threeC, 1u, 5u, 3u);
    tdm_load_2d(vL[bufI], g + 2 * C, 64u, 64u, (unsigned)threeC, 1u, 5u, 3u);
  };

  if (wave == 0) stageQKV(0, 0);
  for (int i = tid; i < 64 * 65; i += 256) (&S[0][0])[i] = 0.f;
  if (tid < 64) ksum[tid] = 0.f;

  for (int c0 = 0; c0 < T; c0 += 64) {
    const int buf = (c0 >> 6) & 1;
    if (wave == 0) __builtin_amdgcn_s_wait_tensorcnt((short)0);
    __syncthreads();  // staged tiles + S/ksum updates visible to all waves
    if (wave == 0 && c0 + 64 < T) stageQKV(buf ^ 1, c0 + 64);
    const float* Q_ = Qs[buf];
    const float* K_ = Ks[buf];
    const float* V_ = Vs[buf];

    // ---- A = Q K^T, causal mask, into LDS ----
#pragma unroll
    for (int tt = 0; tt < 2; ++tt) {
      const int t = wave * 2 + tt;
      const int i0 = (t >> 2) * 16, j0 = (t & 3) * 16;
      v8f acc = (v8f){};
#pragma unroll
      for (int k = 0; k < 64; k += 4) {
        v2f a = *(const v2f*)&Q_[(i0 + l15) * QPITCH + k + 2 * hi];
        v2f bb = *(const v2f*)&K_[(j0 + l15) * QPITCH + k + 2 * hi];
        acc = wmma4(a, bb, acc);
      }
#pragma unroll
      for (int r = 0; r < 8; ++r) {
        const int i = i0 + r + 8 * hi;
        const int j = j0 + l15;
        Am[i][j] = (j <= i) ? acc[r] : 0.f;
      }
    }
    __syncthreads();

    // ---- z[t] = rowsum(Am[t]) + Q[t].ksum_prev ----
    if (tid < 64) {
      float zz = 0.f;
      for (int f = 0; f < 64; ++f)
        zz += Am[tid][f] + Q_[tid * QPITCH + f] * ksum[f];
      zinv[tid] = 1.f / (zz + 1e-6f);
    }
    __syncthreads();

    // ---- O = Q S + Am V, scaled by zinv ----
#pragma unroll
    for (int tt = 0; tt < 2; ++tt) {
      const int t = wave * 2 + tt;
      const int i0 = (t >> 2) * 16, g0 = (t & 3) * 16;
      v8f acc = (v8f){};
#pragma unroll
      for (int k = 0; k < 64; k += 4) {  // Q @ S
        v2f a = *(const v2f*)&Q_[(i0 + l15) * QPITCH + k + 2 * hi];
        v2f bb;
        bb.x = S[k + 2 * hi][g0 + l15];
        bb.y = S[k + 2 * hi + 1][g0 + l15];
        acc = wmma4(a, bb, acc);
      }
#pragma unroll
      for (int k = 0; k < 64; k += 4) {  // Am @ V
        v2f a = *(const v2f*)&Am[i0 + l15][k + 2 * hi];
        v2f bb;
        bb.x = V_[(k + 2 * hi) * QPITCH + g0 + l15];
        bb.y = V_[(k + 2 * hi + 1) * QPITCH + g0 + l15];
        acc = wmma4(a, bb, acc);
      }
#pragma unroll
      for (int r = 0; r < 8; ++r) {
        const int trow = i0 + r + 8 * hi;
        Obase[(size_t)(c0 + trow) * C + g0 + l15] = acc[r] * zinv[trow];
      }
    }
    __syncthreads();  // all reads of S/ksum done before update

    // ---- S += K^T V ----
#pragma unroll
    for (int tt = 0; tt < 2; ++tt) {
      const int t = wave * 2 + tt;
      const int f0 = (t >> 2) * 16, g0 = (t & 3) * 16;
      v8f acc;
#pragma unroll
      for (int r = 0; r < 8; ++r) acc[r] = S[f0 + r + 8 * hi][g0 + l15];
#pragma unroll
      for (int k = 0; k < 64; k += 4) {
        v2f a;  // A = K^T: A[f][t] = K[t][f]
        a.x = K_[(k + 2 * hi) * QPITCH + f0 + l15];
        a.y = K_[(k + 2 * hi + 1) * QPITCH + f0 + l15];
        v2f bb;
        bb.x = V_[(k + 2 * hi) * QPITCH + g0 + l15];
        bb.y = V_[(k + 2 * hi + 1) * QPITCH + g0 + l15];
        acc = wmma4(a, bb, acc);
      }
#pragma unroll
      for (int r = 0; r < 8; ++r) S[f0 + r + 8 * hi][g0 + l15] = acc[r];
    }
    // ---- ksum += column sums of K ----
    if (tid < 64) {
      float s = ksum[tid];
      for (int tr = 0; tr < 64; ++tr) s += K_[tr * QPITCH + tid];
      ksum[tid] = s;
    }
    __syncthreads();
  }
}

// ---------------------------------------------------------------------------
extern "C" void kernel_launch(void* const* d_in, const int* in_sizes, int n_in,
                              void* d_out, int out_size, void* d_ws,
                              size_t ws_size, hipStream_t stream) {
  (void)in_sizes;
  (void)n_in;
  (void)out_size;
  (void)ws_size;
  const float* x = (const float*)d_in[0];
  const float* Wqkv = (const float*)d_in[1];
  const float* bqkv = (const float*)d_in[2];
  const float* gamma_q = (const float*)d_in[3];
  const float* beta_q = (const float*)d_in[4];
  const float* gamma_k = (const float*)d_in[5];
  const float* beta_k = (const float*)d_in[6];
  const float* Wo = (const float*)d_in[7];
  const float* bo = (const float*)d_in[8];
  float* out = (float*)d_out;

  const int B = 2, T = 2048, C = 1024, Hh = 16, D = 64;
  const int M = B * T;       // 4096
  const int threeC = 3 * C;  // 3072

  float* qkv = (float*)d_ws;               // [M, 3C]  48 MB
  float* attn = qkv + (size_t)M * threeC;  // [M, C]   16 MB

  gemm_bias_kernel<<<dim3(M / 128, threeC / 128), 256, 0, stream>>>(
      x, Wqkv, bqkv, qkv, M, threeC, C);
  featuremap_kernel<<<dim3(M, 2), 256, 0, stream>>>(qkv, gamma_q, beta_q,
                                                    gamma_k, beta_k, C, threeC);
  rebased_attn_kernel<<<B * Hh, 256, 0, stream>>>(qkv, attn, T, C, Hh, D);
  gemm_bias_kernel<<<dim3(M / 128, C / 128), 256, 0, stream>>>(attn, Wo, bo,
                                                               out, M, C, C);
}